// AdditiveMul_33509334843954
// MI455X (gfx1250) — compile-verified
//
#include <hip/hip_runtime.h>

typedef float v2f __attribute__((ext_vector_type(2)));
typedef float v8f __attribute__((ext_vector_type(8)));

#define N_Q    2048
#define N_K    2048
#define N_HEAD 8
#define D_HEAD 64
#define KDIM   (N_HEAD * D_HEAD)   // 512 flattened (h,d)

// ---------------------------------------------------------------------------
// Kernel 1: sq / sk via V_WMMA_F32_16X16X4_F32.
//   sq = Q[2048 x 512] @ Wq[512 x 16],  Wq[h*64+d, n] = (n==h) ? a_q[h,d] : 0
//   sk = K[2048 x 512] @ Wk[512 x 16],  Wk[h*64+d, n] = (n==h) ? a_k[h,d] : 0
// One wave (32 lanes) computes a 16x16 output tile, iterating K=512 in steps
// of 4. Only columns n<8 (real heads) are stored.
// A layout (32-bit 16x4, ISA 7.12.2): lanes 0-15 -> M=lane, V0=K0,V1=K1;
// lanes 16-31 -> M=lane-16, V0=K2,V1=K3. B (4x16) mirrors this transposed.
// C/D layout: VGPR v: lanes 0-15 -> M=v, N=lane; lanes 16-31 -> M=v+8.
// ---------------------------------------------------------------------------
__global__ __launch_bounds__(32) void
wmma_row_logits(const float* __restrict__ qm,
                const float* __restrict__ km,
                const float* __restrict__ attn,   // [8][128] = {a_q | a_k}
                float* __restrict__ sq,
                float* __restrict__ sk)
{
    const int lane = threadIdx.x;       // 0..31
    const int half = lane >> 4;         // 0/1
    const int l    = lane & 15;
    const int tile = blockIdx.x;        // 128 tiles of 16 rows
    const int which = blockIdx.y;       // 0 -> sq, 1 -> sk

    const float* __restrict__ src = which ? km : qm;
    float*       __restrict__ dst = which ? sk : sq;
    const int attn_off = which ? D_HEAD : 0;

    const float* arow = src + (size_t)(tile * 16 + l) * KDIM;
    const int n = l;                    // output head column for this lane

    v8f c = {0.f, 0.f, 0.f, 0.f, 0.f, 0.f, 0.f, 0.f};

    for (int k0 = 0; k0 < KDIM; k0 += 4) {
        const int kk = k0 + half * 2;   // this lane's K pair: kk, kk+1
        v2f a;
        a.x = arow[kk + 0];
        a.y = arow[kk + 1];
        // Block-diagonal weight matrix built in registers (selects, no branch).
        const int h = kk >> 6;          // kk and kk+1 share a head (k0 % 4 == 0)
        const float w0 = attn[h * (2 * D_HEAD) + attn_off + ((kk + 0) & 63)];
        const float w1 = attn[h * (2 * D_HEAD) + attn_off + ((kk + 1) & 63)];
        v2f b;
        b.x = (h == n) ? w0 : 0.0f;
        b.y = (h == n) ? w1 : 0.0f;
        c = __builtin_amdgcn_wmma_f32_16x16x4_f32(
                /*neg_a=*/false, a, /*neg_b=*/false, b,
                /*c_mod=*/(short)0, c, /*reuse_a=*/false, /*reuse_b=*/false);
    }

    if (n < N_HEAD) {
        #pragma unroll
        for (int v = 0; v < 8; ++v) {
            const int row = tile * 16 + v + half * 8;
            dst[row * N_HEAD + n] = c[v];
        }
    }
}

// ---------------------------------------------------------------------------
// Kernel 2: the store-bandwidth-bound part.
// One thread per (q,k): out[q,k,h] = softmax_h( relu(sq[q,h] + sk[k,h]) ).
// sq row is uniform per block (scalar-load path); sk loads and out stores are
// two B128 per thread -> 1KB contiguous per wave32.
// ---------------------------------------------------------------------------
__global__ __launch_bounds__(256) void
softmax_stream(const float* __restrict__ sq,
               const float* __restrict__ sk,
               float* __restrict__ out)
{
    const int k = blockIdx.x * 256 + threadIdx.x;   // 0..2047
    const int q = blockIdx.y;                       // 0..2047

    const float4 q0 = ((const float4*)sq)[q * 2 + 0];
    const float4 q1 = ((const float4*)sq)[q * 2 + 1];
    const float4 k0 = ((const float4*)sk)[(size_t)k * 2 + 0];
    const float4 k1 = ((const float4*)sk)[(size_t)k * 2 + 1];

    float v0 = fmaxf(q0.x + k0.x, 0.f);
    float v1 = fmaxf(q0.y + k0.y, 0.f);
    float v2 = fmaxf(q0.z + k0.z, 0.f);
    float v3 = fmaxf(q0.w + k0.w, 0.f);
    float v4 = fmaxf(q1.x + k1.x, 0.f);
    float v5 = fmaxf(q1.y + k1.y, 0.f);
    float v6 = fmaxf(q1.z + k1.z, 0.f);
    float v7 = fmaxf(q1.w + k1.w, 0.f);

    float m = fmaxf(fmaxf(fmaxf(v0, v1), fmaxf(v2, v3)),
                    fmaxf(fmaxf(v4, v5), fmaxf(v6, v7)));

    const float LOG2E = 1.4426950408889634f;
    float e0 = __builtin_amdgcn_exp2f((v0 - m) * LOG2E);
    float e1 = __builtin_amdgcn_exp2f((v1 - m) * LOG2E);
    float e2 = __builtin_amdgcn_exp2f((v2 - m) * LOG2E);
    float e3 = __builtin_amdgcn_exp2f((v3 - m) * LOG2E);
    float e4 = __builtin_amdgcn_exp2f((v4 - m) * LOG2E);
    float e5 = __builtin_amdgcn_exp2f((v5 - m) * LOG2E);
    float e6 = __builtin_amdgcn_exp2f((v6 - m) * LOG2E);
    float e7 = __builtin_amdgcn_exp2f((v7 - m) * LOG2E);

    const float s = ((e0 + e1) + (e2 + e3)) + ((e4 + e5) + (e6 + e7));
    const float r = 1.0f / s;   // compiler emits rcp + NR refine; cost hidden

    float4 o0 = make_float4(e0 * r, e1 * r, e2 * r, e3 * r);
    float4 o1 = make_float4(e4 * r, e5 * r, e6 * r, e7 * r);

    float* p = out + ((size_t)q * N_K + (size_t)k) * N_HEAD;
    *(float4*)(p + 0) = o0;
    *(float4*)(p + 4) = o1;
}

extern "C" void kernel_launch(void* const* d_in, const int* in_sizes, int n_in,
                              void* d_out, int out_size, void* d_ws, size_t ws_size,
                              hipStream_t stream)
{
    (void)in_sizes; (void)n_in; (void)out_size; (void)ws_size;
    const float* q    = (const float*)d_in[0];   // [2048][8][64]
    const float* k    = (const float*)d_in[1];   // [2048][8][64]
    const float* attn = (const float*)d_in[2];   // [1][8][128]
    float* out = (float*)d_out;                  // [2048][2048][8]

    float* sq = (float*)d_ws;                    // [2048][8]
    float* sk = sq + N_Q * N_HEAD;               // [2048][8]  (128KB total ws)

    wmma_row_logits<<<dim3(128, 2), 32, 0, stream>>>(q, k, attn, sq, sk);
    softmax_stream<<<dim3(N_K / 256, N_Q), 256, 0, stream>>>(sq, sk, out);
}